// GLN_34376918237681
// MI455X (gfx1250) — compile-verified
//
#include <hip/hip_runtime.h>
#include <stdint.h>

// Problem constants (match reference)
#define B_    256
#define IN_   512
#define C_    4
#define M_    6
#define S1_   255
#define S2_   63
#define K_    64
#define P_    0.001f
#define LMIN_ (-6.906754778648554f)
#define LMAX_ ( 6.906754778648554f)
#define LR_   0.001f
#define WCLIP_ 5.0f

typedef float v2f  __attribute__((ext_vector_type(2)));
typedef float v8f  __attribute__((ext_vector_type(8)));
typedef unsigned int u32x4 __attribute__((ext_vector_type(4)));
typedef int   i32x4 __attribute__((ext_vector_type(4)));
typedef int   i32x8 __attribute__((ext_vector_type(8)));

// ---------------------------------------------------------------------------
// l0[b,i] = log(xb/(1-xb)), xb = clip(x, P, 1-P); column 0 forced to 0.
// ---------------------------------------------------------------------------
__global__ void k_l0(const float* __restrict__ x, float* __restrict__ l0) {
  int t = blockIdx.x * blockDim.x + threadIdx.x;
  if (t >= B_ * IN_) return;
  float xb = fminf(fmaxf(x[t], P_), 1.0f - P_);
  float v = logf(xb / (1.0f - xb));
  if ((t & (IN_ - 1)) == 0) v = 0.0f;       // BASE_BIAS
  l0[t] = v;
}

// ---------------------------------------------------------------------------
// dist[r,b] = sum_k A[r,k] * x[b,k]  via V_WMMA_F32_16X16X4_F32.
// Block = 128 threads = 4 waves; block tile = 64 rows x 16 batch cols.
// The 16x512 f32 x-tile (32KB) is DMA'd to LDS by the Tensor Data Mover.
// dist buffer rows are padded to a multiple of 64 so stores are unguarded.
// ---------------------------------------------------------------------------
__global__ __launch_bounds__(128)
void k_dist_wmma(const float* __restrict__ A, const float* __restrict__ x,
                 float* __restrict__ dist, int R) {
  __shared__ float xs[16 * IN_];            // 32 KB staging of 16 rows of x
  const int ct0  = blockIdx.y * 16;         // batch-column tile origin
  const int rt0  = blockIdx.x * 64;         // row tile origin
  const int lane = threadIdx.x & 31;
  const int wv   = threadIdx.x >> 5;

  // ---- TDM: stage x[ct0 .. ct0+15][0..511] into LDS (wave 0 issues) ----
  if (threadIdx.x < 32) {
    uint64_t ga  = (uint64_t)(uintptr_t)(x + (size_t)ct0 * IN_);
    uint32_t lds = (uint32_t)(uintptr_t)(&xs[0]);   // low 32b of flat = LDS addr
    u32x4 g0;
    g0.x = 1u;                                      // count=1, user descriptor
    g0.y = lds;                                     // lds_addr
    g0.z = (uint32_t)ga;                            // global_addr[31:0]
    g0.w = (uint32_t)(ga >> 32) | (2u << 30);       // global_addr[56:32] | type=2
    i32x8 g1;
    g1[0] = 0x00020000;                             // data_size = 4 bytes
    g1[1] = (int)(IN_ << 16);                       // tensor_dim0[15:0]  -> bits 63:48
    g1[2] = (int)(B_ << 16);                        // tensor_dim1[15:0]  -> bits 95:80
    g1[3] = (int)(IN_ << 16);                       // tile_dim0 = 512    -> bits 127:112
    g1[4] = 16;                                     // tile_dim1 = 16 rows
    g1[5] = IN_;                                    // tensor_dim0_stride = 512
    g1[6] = 0;
    g1[7] = 0;
    i32x4 gz = {0, 0, 0, 0};
#if defined(__clang_major__) && (__clang_major__ >= 23)
    i32x8 gz8 = {0, 0, 0, 0, 0, 0, 0, 0};
    __builtin_amdgcn_tensor_load_to_lds(g0, g1, gz, gz, gz8, 0);
#else
    __builtin_amdgcn_tensor_load_to_lds(g0, g1, gz, gz, 0);
#endif
    __builtin_amdgcn_s_wait_tensorcnt(0);
  }
  __syncthreads();

  // ---- A-fragment addressing (ISA 7.12.2: 32-bit A 16x4) ----
  int arow = rt0 + wv * 16 + (lane & 15);
  if (arow >= R) arow = R - 1;                      // clamp reads on partial tiles
  const int koff = (lane >> 4) * 2;                 // lanes 16-31 hold K+2,K+3
  const float* ap = A + (size_t)arow * IN_ + koff;
  const float* bp = &xs[(size_t)(lane & 15) * IN_ + koff];
  __builtin_prefetch(ap, 0, 1);                     // global_prefetch_b8

  v8f acc = {};
#pragma unroll 4
  for (int kk = 0; kk < IN_; kk += 4) {
    v2f a; a.x = ap[kk]; a.y = ap[kk + 1];
    v2f b; b.x = bp[kk]; b.y = bp[kk + 1];
    acc = __builtin_amdgcn_wmma_f32_16x16x4_f32(false, a, false, b,
                                                (short)0, acc, false, false);
  }

  // ---- store C/D fragment: VGPR g holds rows {g, g+8} per half-wave ----
  const int col   = ct0 + (lane & 15);
  const int rbase = rt0 + wv * 16 + ((lane >> 4) << 3);
#pragma unroll
  for (int g = 0; g < 8; ++g)
    dist[(size_t)(rbase + g) * B_ + col] = acc[g];
}

// ---------------------------------------------------------------------------
// idx[c,s,b] = sum_m (dist[(cs*M+m), b] > 0) << m
// ---------------------------------------------------------------------------
__global__ void k_pack(const float* __restrict__ dist, int* __restrict__ idx, int S) {
  int t = blockIdx.x * blockDim.x + threadIdx.x;
  if (t >= C_ * S * B_) return;
  int b  = t % B_;
  int cs = t / B_;
  const float* dp = dist + (size_t)cs * M_ * B_ + b;
  int v = 0;
#pragma unroll
  for (int m = 0; m < M_; ++m)
    v |= (dp[(size_t)m * B_] > 0.0f) ? (1 << m) : 0;
  idx[t] = v;
}

// ---------------------------------------------------------------------------
// out_t[b,0,c] = bias[c]
// ---------------------------------------------------------------------------
__global__ void k_biasfill(const float* __restrict__ bias,
                           float* __restrict__ out_t, int Sp1) {
  int t = blockIdx.x * blockDim.x + threadIdx.x;
  if (t >= B_ * C_) return;
  int b = t / C_, c = t % C_;
  out_t[((size_t)b * Sp1) * C_ + c] = bias[c];
}

// ---------------------------------------------------------------------------
// Forward: out_t[b, s+off, c] = clip( dot(w[c,s,idx], lg[b,:,c]), LMIN, LMAX )
// lgStrideC==0 -> lg is (B,I) shared across c (layer 1); else lg is (B,I,C).
// ---------------------------------------------------------------------------
__global__ void k_forward(const float* __restrict__ w, const int* __restrict__ idx,
                          const float* __restrict__ lg, int lgStrideC,
                          float* __restrict__ out_t, int S, int I, int off) {
  int t = blockIdx.x * blockDim.x + threadIdx.x;
  if (t >= C_ * S * B_) return;
  int b  = t % B_;
  int cs = t / B_;
  int c  = cs / S, s = cs % S;
  int k  = idx[t];
  const float* wr = w + ((size_t)cs * K_ + k) * I;
  float acc = 0.0f;
  if (lgStrideC == 0) {
    const float4* w4 = (const float4*)wr;
    const float4* g4 = (const float4*)(lg + (size_t)b * I);
    for (int i = 0; i < I / 4; ++i) {
      float4 wv = w4[i], gv = g4[i];
      acc += wv.x * gv.x + wv.y * gv.y + wv.z * gv.z + wv.w * gv.w;
    }
  } else {
    const float* gp = lg + (size_t)b * I * C_ + c;
    for (int i = 0; i < I; ++i)
      acc += wr[i] * gp[(size_t)i * C_];
  }
  float v = fminf(fmaxf(acc, LMIN_), LMAX_);
  out_t[((size_t)b * (S + off) + (s + off)) * C_ + c] = v;
}

// ---------------------------------------------------------------------------
// lastb[c,s,k] = largest b with idx[c,s,b]==k, else -1 (JAX set: last wins).
// One thread per (c,s), serial over b -> deterministic.
// ---------------------------------------------------------------------------
__global__ void k_lastb(const int* __restrict__ idx, int* __restrict__ lastb, int S) {
  int cs = blockIdx.x * blockDim.x + threadIdx.x;
  if (cs >= C_ * S) return;
  int* lb = lastb + (size_t)cs * K_;
  for (int k = 0; k < K_; ++k) lb[k] = -1;
  const int* ip = idx + (size_t)cs * B_;
  for (int b = 0; b < B_; ++b) lb[ip[b]] = b;
}

// ---------------------------------------------------------------------------
// Fused copy+update: nw[c,s,k,:] = (lastb<0) ? w_row
//   : clip( w_row - LR*(sigmoid(out)-tgt)*lg[b,:,c], +-WCLIP ).
// Block per (c,s,k); thread handles a float4 of the I-length row.
// Exactly the HBM floor: one read + one write per weight element.
// ---------------------------------------------------------------------------
__global__ __launch_bounds__(128)
void k_apply(const float* __restrict__ w, const int* __restrict__ lastb,
             const float* __restrict__ out_this, const float* __restrict__ lg,
             int lgStrideC, const int* __restrict__ target,
             float* __restrict__ nw, int S, int I, int off) {
  int csk = blockIdx.x;
  int cs  = csk / K_;
  int c   = cs / S, s = cs % S;
  int i0  = threadIdx.x * 4;
  if (i0 >= I) return;
  size_t base = (size_t)csk * I + i0;               // ((cs*K)+k)*I + i0
  const float4 wv = *(const float4*)(w + base);
  float4 r;
  int b = lastb[csk];
  if (b < 0) {
    r = wv;
  } else {
    float o    = out_this[((size_t)b * (S + off) + (s + off)) * C_ + c];
    float sg   = 1.0f / (1.0f + expf(-o));
    float lrd  = LR_ * (sg - ((target[b] == c) ? 1.0f : 0.0f));
    float e0, e1, e2, e3;
    if (lgStrideC == 0) {
      const float4 gv = *(const float4*)(lg + (size_t)b * I + i0);
      e0 = gv.x; e1 = gv.y; e2 = gv.z; e3 = gv.w;
    } else {
      const float* gp = lg + ((size_t)b * I + i0) * C_ + c;
      e0 = gp[0]; e1 = gp[C_]; e2 = gp[2 * C_]; e3 = gp[3 * C_];
    }
    r.x = fminf(fmaxf(wv.x - lrd * e0, -WCLIP_), WCLIP_);
    r.y = fminf(fmaxf(wv.y - lrd * e1, -WCLIP_), WCLIP_);
    r.z = fminf(fmaxf(wv.z - lrd * e2, -WCLIP_), WCLIP_);
    r.w = fminf(fmaxf(wv.w - lrd * e3, -WCLIP_), WCLIP_);
  }
  *(float4*)(nw + base) = r;
}

// ---------------------------------------------------------------------------
extern "C" void kernel_launch(void* const* d_in, const int* in_sizes, int n_in,
                              void* d_out, int out_size, void* d_ws, size_t ws_size,
                              hipStream_t stream) {
  (void)in_sizes; (void)n_in; (void)out_size; (void)ws_size;
  const float* x      = (const float*)d_in[0];
  const int*   target = (const int*)  d_in[1];
  const float* cmaps1 = (const float*)d_in[2];
  const float* w1     = (const float*)d_in[3];
  const float* bias1  = (const float*)d_in[4];
  const float* cmaps2 = (const float*)d_in[5];
  const float* w2     = (const float*)d_in[6];
  const float* bias2  = (const float*)d_in[7];
  const float* cmaps3 = (const float*)d_in[8];
  const float* w3     = (const float*)d_in[9];

  float* out    = (float*)d_out;
  float* logits = out;                                       // (B,C)
  float* nw1    = out + (size_t)B_ * C_;                     // (C,S1,K,IN)
  float* nw2    = nw1 + (size_t)C_ * S1_ * K_ * IN_;         // (C,S2,K,S1+1)
  float* nw3    = nw2 + (size_t)C_ * S2_ * K_ * (S1_ + 1);   // (C,1,K,S2+1)

  // Workspace carve-up (~11.5 MB), 256B aligned chunks.
  char* p = (char*)d_ws;
  auto carve = [&](size_t bytes) -> void* {
    void* r = (void*)p;
    p += (bytes + 255) & ~(size_t)255;
    return r;
  };
  float* l0    = (float*)carve(sizeof(float) * B_ * IN_);
  float* dist1 = (float*)carve(sizeof(float) * 6144 * B_);   // R1=6120 pad->6144
  float* dist2 = (float*)carve(sizeof(float) * 1536 * B_);   // R2=1512 pad->1536
  float* dist3 = (float*)carve(sizeof(float) * 64 * B_);     // R3=24   pad->64
  int*   idx1  = (int*)  carve(sizeof(int) * C_ * S1_ * B_);
  int*   idx2  = (int*)  carve(sizeof(int) * C_ * S2_ * B_);
  int*   idx3  = (int*)  carve(sizeof(int) * C_ * 1 * B_);
  float* ot1   = (float*)carve(sizeof(float) * B_ * (S1_ + 1) * C_);
  float* ot2   = (float*)carve(sizeof(float) * B_ * (S2_ + 1) * C_);
  int*   lb1   = (int*)  carve(sizeof(int) * C_ * S1_ * K_);
  int*   lb2   = (int*)  carve(sizeof(int) * C_ * S2_ * K_);
  int*   lb3   = (int*)  carve(sizeof(int) * C_ * 1 * K_);

  const int R1 = C_ * S1_ * M_, R2 = C_ * S2_ * M_, R3 = C_ * 1 * M_;

  // Base logits
  k_l0<<<(B_ * IN_ + 255) / 256, 256, 0, stream>>>(x, l0);

  // Three context GEMMs (independent; shared B-matrix x) + bit-pack.
  k_dist_wmma<<<dim3((R1 + 63) / 64, B_ / 16), 128, 0, stream>>>(cmaps1, x, dist1, R1);
  k_dist_wmma<<<dim3((R2 + 63) / 64, B_ / 16), 128, 0, stream>>>(cmaps2, x, dist2, R2);
  k_dist_wmma<<<dim3((R3 + 63) / 64, B_ / 16), 128, 0, stream>>>(cmaps3, x, dist3, R3);
  k_pack<<<(C_ * S1_ * B_ + 255) / 256, 256, 0, stream>>>(dist1, idx1, S1_);
  k_pack<<<(C_ * S2_ * B_ + 255) / 256, 256, 0, stream>>>(dist2, idx2, S2_);
  k_pack<<<(C_ * 1 * B_ + 255) / 256, 256, 0, stream>>>(dist3, idx3, 1);

  // Layer 1 (lg = l0, shared across classes)
  k_biasfill<<<(B_ * C_ + 255) / 256, 256, 0, stream>>>(bias1, ot1, S1_ + 1);
  k_forward<<<(C_ * S1_ * B_ + 127) / 128, 128, 0, stream>>>(w1, idx1, l0, 0, ot1, S1_, IN_, 1);
  k_lastb<<<(C_ * S1_ + 127) / 128, 128, 0, stream>>>(idx1, lb1, S1_);
  k_apply<<<C_ * S1_ * K_, 128, 0, stream>>>(w1, lb1, ot1, l0, 0, target, nw1, S1_, IN_, 1);

  // Layer 2 (lg = ot1, (B,S1+1,C))
  k_biasfill<<<(B_ * C_ + 255) / 256, 256, 0, stream>>>(bias2, ot2, S2_ + 1);
  k_forward<<<(C_ * S2_ * B_ + 127) / 128, 128, 0, stream>>>(w2, idx2, ot1, C_, ot2, S2_, S1_ + 1, 1);
  k_lastb<<<(C_ * S2_ + 127) / 128, 128, 0, stream>>>(idx2, lb2, S2_);
  k_apply<<<C_ * S2_ * K_, 128, 0, stream>>>(w2, lb2, ot2, ot1, C_, target, nw2, S2_, S1_ + 1, 1);

  // Layer 3 (no bias): (B,1,C) layout == (B,C) logits, write d_out directly.
  k_forward<<<(C_ * 1 * B_ + 127) / 128, 128, 0, stream>>>(w3, idx3, ot2, C_, logits, 1, S2_ + 1, 0);
  k_lastb<<<(C_ * 1 + 127) / 128, 128, 0, stream>>>(idx3, lb3, 1);
  k_apply<<<C_ * 1 * K_, 128, 0, stream>>>(w3, lb3, logits, ot2, C_, target, nw3, 1, S2_ + 1, 0);
}